// GINBlock_3805341024429
// MI455X (gfx1250) — compile-verified
//
#include <hip/hip_runtime.h>

typedef __attribute__((ext_vector_type(2))) float v2f;
typedef __attribute__((ext_vector_type(8))) float v8f;

// ---------------------------------------------------------------- copy h = x
__global__ void copy_f4(float4* __restrict__ dst, const float4* __restrict__ src, int n) {
    int i = blockIdx.x * blockDim.x + threadIdx.x;
    if (i < n) dst[i] = src[i];
}

// ------------------------------------------- GIN scatter: h[dst] += x[src]
// 32 threads per edge, each thread handles 2 channels (2 f32 atomics).
__global__ void edge_scatter(float* __restrict__ h, const float* __restrict__ x,
                             const long long* __restrict__ ei, int E) {
    long long gid = (long long)blockIdx.x * blockDim.x + threadIdx.x;
    long long e = gid >> 5;
    if (e >= E) return;
    int c = (int)(gid & 31) * 2;
    long long s = ei[e];
    long long d = ei[(long long)E + e];
    atomicAdd(&h[d * 64 + c],     x[s * 64 + c]);
    atomicAdd(&h[d * 64 + c + 1], x[s * 64 + c + 1]);
}

// ------------------------- fused  out = ReLU(LayerNorm(Hin @ W^T))  via WMMA
// Block = 256 threads = 8 waves; each wave owns 16 rows (16x64 output tile).
// A (16x4 f32) per ISA layout: lane M=lane%16, VGPR0/1 = K = 2*laneHi + {0,1}.
// B (4x16 f32) mirrored: lane n=lane%16 holds W[n][k0+2*laneHi +{0,1}] from LDS.
__global__ __launch_bounds__(256)
void gemm_ln_relu(float* __restrict__ out, const float* __restrict__ Hin,
                  const float* __restrict__ W, const float* __restrict__ lnw,
                  const float* __restrict__ lnb, int N) {
    __shared__ float sW[64 * 64];        // 16 KB, W row-major (W[n][k])
    __shared__ float sT[8 * 16 * 64];    // 32 KB, per-wave 16x64 result tiles

    int tid = threadIdx.x;
    {   // cooperative W load (4096 floats = 1024 float4)
        const float4* src = (const float4*)W;
        float4* dst = (float4*)sW;
        #pragma unroll
        for (int i = 0; i < 4; ++i) dst[tid + i * 256] = src[tid + i * 256];
    }
    __syncthreads();

    int wave   = tid >> 5;
    int lane   = tid & 31;
    int laneLo = lane & 15;
    int laneHi = lane >> 4;
    int r  = blockIdx.x * 128 + wave * 16 + laneLo;
    int rc = r < N ? r : N - 1;          // clamp: keep EXEC all-1s for WMMA

    // Preload all 16 A fragments (K = 0..63)
    v2f a[16];
    const float* ap = Hin + (size_t)rc * 64 + 2 * laneHi;
    #pragma unroll
    for (int kb = 0; kb < 16; ++kb) a[kb] = *(const v2f*)(ap + 4 * kb);

    float* tile = sT + wave * (16 * 64);
    #pragma unroll
    for (int nb = 0; nb < 4; ++nb) {
        v8f acc = {0.f, 0.f, 0.f, 0.f, 0.f, 0.f, 0.f, 0.f};
        const float* bp = sW + (nb * 16 + laneLo) * 64 + 2 * laneHi;
        #pragma unroll
        for (int kb = 0; kb < 16; ++kb) {
            v2f b = *(const v2f*)(bp + 4 * kb);
            acc = __builtin_amdgcn_wmma_f32_16x16x4_f32(
                false, a[kb], false, b, (short)0, acc, false, false);
        }
        // D layout: VGPR j -> row M = j + 8*laneHi, col N = laneLo (+16*nb)
        #pragma unroll
        for (int j = 0; j < 8; ++j)
            tile[(j + 8 * laneHi) * 64 + laneLo + 16 * nb] = acc[j];
    }
    __syncthreads();

    // LayerNorm + ReLU: 2 lanes per row (laneLo = row, laneHi = column half)
    const float* rp = tile + laneLo * 64 + laneHi * 32;
    float s1 = 0.f, s2 = 0.f;
    #pragma unroll
    for (int i = 0; i < 32; ++i) { float f = rp[i]; s1 += f; s2 += f * f; }
    s1 += __shfl_xor(s1, 16, 32);
    s2 += __shfl_xor(s2, 16, 32);
    float mu   = s1 * (1.0f / 64.0f);
    float var  = s2 * (1.0f / 64.0f) - mu * mu;
    float rstd = rsqrtf(var + 1e-5f);
    if (r < N) {
        float* op = out + (size_t)r * 64 + laneHi * 32;
        const float* wv = lnw + laneHi * 32;
        const float* bv = lnb + laneHi * 32;
        #pragma unroll
        for (int i = 0; i < 32; ++i) {
            float f = (rp[i] - mu) * rstd * wv[i] + bv[i];
            op[i] = f > 0.f ? f : 0.f;
        }
    }
}

// ------------------------------------------------------------ GraphNorm pass
__global__ void zero128(float* p) { if (threadIdx.x < 128) p[threadIdx.x] = 0.f; }

__global__ void col_sum(float* __restrict__ S, const float* __restrict__ h, int N) {
    __shared__ float red[256];
    int c = threadIdx.x & 63;
    int g = threadIdx.x >> 6;           // 0..3 rows in parallel per channel
    float acc = 0.f;
    for (int r = blockIdx.x * 4 + g; r < N; r += gridDim.x * 4)
        acc += h[(size_t)r * 64 + c];
    red[threadIdx.x] = acc;
    __syncthreads();
    if (threadIdx.x < 64)
        atomicAdd(&S[threadIdx.x],
                  red[threadIdx.x] + red[threadIdx.x + 64] +
                  red[threadIdx.x + 128] + red[threadIdx.x + 192]);
}

__global__ void col_var(float* __restrict__ V, const float* __restrict__ h,
                        const float* __restrict__ S, const float* __restrict__ alpha, int N) {
    __shared__ float red[256];
    int c = threadIdx.x & 63;
    int g = threadIdx.x >> 6;
    float am = alpha[c] * S[c] * (1.0f / (float)N);   // alpha * mean
    float acc = 0.f;
    for (int r = blockIdx.x * 4 + g; r < N; r += gridDim.x * 4) {
        float d = h[(size_t)r * 64 + c] - am;
        acc += d * d;
    }
    red[threadIdx.x] = acc;
    __syncthreads();
    if (threadIdx.x < 64)
        atomicAdd(&V[threadIdx.x],
                  red[threadIdx.x] + red[threadIdx.x + 64] +
                  red[threadIdx.x + 128] + red[threadIdx.x + 192]);
}

__global__ void gn_final(float* __restrict__ out, const float* __restrict__ h,
                         const float* __restrict__ S, const float* __restrict__ V,
                         const float* __restrict__ gw, const float* __restrict__ gb,
                         const float* __restrict__ ga, int N) {
    int i = blockIdx.x * blockDim.x + threadIdx.x;
    if (i >= N * 64) return;
    int c = i & 63;
    float invN = 1.0f / (float)N;
    float mean = S[c] * invN;
    float var  = V[c] * invN;
    float o = h[i] - ga[c] * mean;
    out[i] = gw[c] * o * rsqrtf(var + 1e-5f) + gb[c];
}

// --------------------------------------------------------------------- launch
extern "C" void kernel_launch(void* const* d_in, const int* in_sizes, int n_in,
                              void* d_out, int out_size, void* d_ws, size_t ws_size,
                              hipStream_t stream) {
    const float*      x    = (const float*)d_in[0];
    const long long*  ei   = (const long long*)d_in[1];
    const float*      W0   = (const float*)d_in[2];
    const float*      ln0w = (const float*)d_in[3];
    const float*      ln0b = (const float*)d_in[4];
    const float*      W1   = (const float*)d_in[5];
    const float*      ln1w = (const float*)d_in[6];
    const float*      ln1b = (const float*)d_in[7];
    const float*      gnw  = (const float*)d_in[8];
    const float*      gnb  = (const float*)d_in[9];
    const float*      gna  = (const float*)d_in[10];

    int N = in_sizes[0] / 64;
    int E = in_sizes[1] / 2;

    float* S    = (float*)d_ws;               // [64] channel sums
    float* V    = S + 64;                     // [64] channel sq-sums
    float* hbuf = S + 256;                    // [N*64] ping
    float* h1   = hbuf + (size_t)N * 64;      // [N*64] pong
    float* out  = (float*)d_out;

    // 1) h = x
    int nv4 = (N * 64) / 4;
    copy_f4<<<(nv4 + 255) / 256, 256, 0, stream>>>((float4*)hbuf, (const float4*)x, nv4);
    // 2) h[dst] += x[src]
    long long eth = (long long)E * 32;
    edge_scatter<<<(int)((eth + 255) / 256), 256, 0, stream>>>(hbuf, x, ei, E);
    // 3) MLP layer 0 and 1 (fused GEMM + LN + ReLU, WMMA f32)
    int gb = (N + 127) / 128;
    gemm_ln_relu<<<gb, 256, 0, stream>>>(h1, hbuf, W0, ln0w, ln0b, N);
    gemm_ln_relu<<<gb, 256, 0, stream>>>(hbuf, h1, W1, ln1w, ln1b, N);
    // 4) GraphNorm
    zero128<<<1, 128, 0, stream>>>(S);
    col_sum<<<128, 256, 0, stream>>>(S, hbuf, N);
    col_var<<<128, 256, 0, stream>>>(V, hbuf, S, gna, N);
    int tot = N * 64;
    gn_final<<<(tot + 255) / 256, 256, 0, stream>>>(out, hbuf, S, V, gnw, gnb, gna, N);
}